// Kmeans_22153441312892
// MI455X (gfx1250) — compile-verified
//
#include <hip/hip_runtime.h>

#define NPTS  2097152
#define DIM   32
#define KCLUS 64
#define ITERS 11      // ITERATIONS + 1 scan steps in the reference

typedef __attribute__((ext_vector_type(2))) float v2f;
typedef __attribute__((ext_vector_type(8))) float v8f;

// Workspace layout (floats):
//   [0,    2048)  current centroids [64][32]
//   [2048, 2112)  c2[k] = ||centroid_k||^2
//   [2112, 4160)  sums  [64][32]  (scatter-add accumulator)
//   [4160, 4224)  counts[64]
#define WS_CENT 0
#define WS_C2   2048
#define WS_SUMS 2112
#define WS_CNT  4160

#define ABLOCKS 512
#define WPB 8   // waves per 256-thread block (wave32)

// ---------------------------------------------------------------------------
// Init: centroids <- init_centroids, compute c2, zero accumulators.
// Single block; __syncthreads orders centroid writes before c2 reads.
// ---------------------------------------------------------------------------
__global__ __launch_bounds__(256) void kinit(const float* __restrict__ initc,
                                             float* __restrict__ ws) {
  float* cent = ws + WS_CENT;
  float* c2   = ws + WS_C2;
  float* sums = ws + WS_SUMS;
  float* cnts = ws + WS_CNT;
  int t = threadIdx.x;
  for (int i = t; i < 2048; i += 256) { cent[i] = initc[i]; sums[i] = 0.0f; }
  if (t < 64) cnts[t] = 0.0f;
  __syncthreads();
  if (t < 64) {
    float s = 0.0f;
    for (int d = 0; d < DIM; ++d) { float v = cent[t * DIM + d]; s += v * v; }
    c2[t] = s;
  }
}

// ---------------------------------------------------------------------------
// Fused assignment + accumulation.
// Each wave owns 16-point tiles; cross[k,n] via V_WMMA_F32_16X16X4_F32:
//   M = 64 centroids (4 tiles of 16), N = 16 points, K-dim = 32 (8 chunks of 4).
// score = c2[k] - 2*cross (x2 term is constant per point -> drop for argmin).
// Per-block LDS scatter-add, one global atomic flush at block end.
// ---------------------------------------------------------------------------
__global__ __launch_bounds__(256) void kassign(const float* __restrict__ X,
                                               float* ws,
                                               int* __restrict__ assign_out) {
  const float* cent = ws + WS_CENT;
  const float* c2g  = ws + WS_C2;
  float* gsums = ws + WS_SUMS;
  float* gcnts = ws + WS_CNT;

  __shared__ float lsums[KCLUS * DIM];
  __shared__ float lcnts[KCLUS];

  const int tid = threadIdx.x;
  for (int i = tid; i < KCLUS * DIM; i += 256) lsums[i] = 0.0f;
  if (tid < KCLUS) lcnts[tid] = 0.0f;
  __syncthreads();

  const int lane = tid & 31;
  const int wave = tid >> 5;
  const int l16  = lane & 15;   // point / centroid row within tile
  const int h    = lane >> 4;   // selects K half (A/B) and M half (C/D)

  // Hoist A fragments (centroids) and c2 into registers: constant all kernel.
  // A 16x4 f32 layout: lanes 0-15 -> (K=kk, K=kk+1); lanes 16-31 -> (kk+2, kk+3).
  v2f   afrag[4][8];
  float c2r[4][8];
  for (int mt = 0; mt < 4; ++mt) {
    const float* crow = cent + (mt * 16 + l16) * DIM;
    for (int kc = 0; kc < 8; ++kc)
      afrag[mt][kc] = *(const v2f*)(crow + kc * 4 + 2 * h);
    // C/D layout: vgpr j, lane-half h -> centroid m = mt*16 + j + 8*h
    for (int j = 0; j < 8; ++j)
      c2r[mt][j] = c2g[mt * 16 + j + 8 * h];
  }

  const int gw     = blockIdx.x * WPB + wave;
  const int nwaves = gridDim.x * WPB;
  const int ntiles = NPTS / 16;

  for (int tile = gw; tile < ntiles; tile += nwaves) {
    const int n0 = tile * 16;
    const float* xrow = X + (long)(n0 + l16) * DIM;

    // Prefetch next tile's slice while WMMAs run (global_prefetch_b8).
    if (tile + nwaves < ntiles)
      __builtin_prefetch(X + (long)(tile + nwaves) * 16 * DIM + (long)l16 * DIM, 0, 1);

    v8f acc[4];
    for (int mt = 0; mt < 4; ++mt)
      for (int j = 0; j < 8; ++j) acc[mt][j] = 0.0f;

    for (int kc = 0; kc < 8; ++kc) {
      // B 4x16 f32: lane = point column; lane-half selects K pair (mirrors A).
      v2f b = *(const v2f*)(xrow + kc * 4 + 2 * h);
      for (int mt = 0; mt < 4; ++mt) {
        acc[mt] = __builtin_amdgcn_wmma_f32_16x16x4_f32(
            false, afrag[mt][kc], false, b, (short)0, acc[mt], false, false);
      }
    }

    // Local argmin over this lane's 32 centroids (first-min tie-break).
    float best = 3.402823466e38f;
    int   bidx = 0;
    for (int mt = 0; mt < 4; ++mt)
      for (int j = 0; j < 8; ++j) {
        float s = c2r[mt][j] - 2.0f * acc[mt][j];
        int   m = mt * 16 + j + 8 * h;
        if (s < best) { best = s; bidx = m; }
      }
    // Combine lane pair (lane ^ 16): both lanes end with the full argmin.
    float obest = __shfl_xor(best, 16, 32);
    int   oidx  = __shfl_xor(bidx, 16, 32);
    if (obest < best || (obest == best && oidx < bidx)) { best = obest; bidx = oidx; }

    if (h == 0) assign_out[n0 + l16] = bidx;

    // Scatter-add point into LDS accumulator: lane pair splits the 32 dims.
    const float* xr   = X + (long)(n0 + l16) * DIM + 16 * h;
    float*       srow = lsums + bidx * DIM + 16 * h;
    for (int i = 0; i < 16; ++i) atomicAdd(&srow[i], xr[i]);  // ds_add_f32
    if (h == 0) atomicAdd(&lcnts[bidx], 1.0f);
  }

  __syncthreads();
  for (int i = tid; i < KCLUS * DIM; i += 256) atomicAdd(&gsums[i], lsums[i]);
  if (tid < KCLUS) atomicAdd(&gcnts[tid], lcnts[tid]);
}

// ---------------------------------------------------------------------------
// Centroid update: mean (or keep old if empty), recompute c2, re-zero accums.
// ---------------------------------------------------------------------------
__global__ __launch_bounds__(256) void kupdate(float* __restrict__ ws) {
  float* cent = ws + WS_CENT;
  float* c2   = ws + WS_C2;
  float* sums = ws + WS_SUMS;
  float* cnts = ws + WS_CNT;
  int t = threadIdx.x;
  for (int i = t; i < 2048; i += 256) {
    float c = cnts[i >> 5];
    cent[i] = (c > 0.0f) ? (sums[i] / c) : cent[i];
  }
  __syncthreads();
  if (t < 64) {
    float s = 0.0f;
    for (int d = 0; d < DIM; ++d) { float v = cent[t * DIM + d]; s += v * v; }
    c2[t] = s;
  }
  __syncthreads();
  for (int i = t; i < 2048; i += 256) sums[i] = 0.0f;
  if (t < 64) cnts[t] = 0.0f;
}

__global__ __launch_bounds__(256) void kfinal(const float* __restrict__ ws,
                                              float* __restrict__ outc) {
  int i = blockIdx.x * 256 + threadIdx.x;
  if (i < 2048) outc[i] = ws[WS_CENT + i];
}

// ---------------------------------------------------------------------------
extern "C" void kernel_launch(void* const* d_in, const int* in_sizes, int n_in,
                              void* d_out, int out_size, void* d_ws, size_t ws_size,
                              hipStream_t stream) {
  const float* X     = (const float*)d_in[0];  // [N, 32] points
  const float* initc = (const float*)d_in[1];  // [64, 32] init centroids
  float* ws = (float*)d_ws;

  int*   assign_out = (int*)d_out;             // first N elems: int32 assignments
  float* cent_out   = (float*)d_out + NPTS;    // next 2048: final centroids

  kinit<<<1, 256, 0, stream>>>(initc, ws);
  for (int it = 0; it < ITERS; ++it) {
    kassign<<<ABLOCKS, 256, 0, stream>>>(X, ws, assign_out);
    kupdate<<<1, 256, 0, stream>>>(ws);
  }
  kfinal<<<8, 256, 0, stream>>>(ws, cent_out);
}